// VectorQuantizer_72851235275123
// MI455X (gfx1250) — compile-verified
//
#include <hip/hip_runtime.h>

typedef __attribute__((ext_vector_type(2))) float v2f;
typedef __attribute__((ext_vector_type(8))) float v8f;
typedef __attribute__((address_space(3))) const float lds_cfloat;

#define NUM_LATENTS 1024
#define LATENT_DIM  64
#define BATCH       32
#define SEQ         4096
#define N_TOTAL     (BATCH * SEQ)               // 131072 points
#define S_OUT       (BATCH * LATENT_DIM * SEQ)  // elems per (b,c,l) output section

#define CHUNK_K      256                         // codebook rows per LDS chunk
#define NCHUNK       (NUM_LATENTS / CHUNK_K)     // 4
#define CHUNK_FLOATS (CHUNK_K * LATENT_DIM)      // 16384 floats = 64 KB
#define ROWS_PER_WAVE   32
#define WAVES_PER_BLOCK 8
#define ROWS_PER_BLOCK  (ROWS_PER_WAVE * WAVES_PER_BLOCK)  // 256

// ---------------------------------------------------------------------------
// Pre-kernel: cbh[k] = 0.5 * ||codebook[k]||^2   (folded into WMMA C-init)
// ---------------------------------------------------------------------------
__global__ __launch_bounds__(1024) void vq_cbh_kernel(const float* __restrict__ cb,
                                                      float* __restrict__ cbh) {
    int k = blockIdx.x * blockDim.x + threadIdx.x;
    if (k < NUM_LATENTS) {
        const float* row = cb + (size_t)k * LATENT_DIM;
        float s = 0.f;
#pragma unroll
        for (int c = 0; c < LATENT_DIM; ++c) s += row[c] * row[c];
        cbh[k] = 0.5f * s;
    }
}

// ---------------------------------------------------------------------------
// Async DMA of one 64KB codebook chunk into LDS (CDNA5 ASYNCcnt path).
// 256 threads x 16 iters x 16B = 64KB. Per-wave completion via s_wait_asynccnt.
// ---------------------------------------------------------------------------
__device__ __forceinline__ void issue_chunk_async(const float* __restrict__ gsrc,
                                                  const float* lbase, int tid) {
    const unsigned lds0 = (unsigned)(size_t)(lds_cfloat*)lbase;  // LDS byte offset
#pragma unroll
    for (int it = 0; it < 16; ++it) {
        const unsigned byteoff = (unsigned)((it * 256 + tid) * 16);
        const unsigned ldst = lds0 + byteoff;
        asm volatile("global_load_async_to_lds_b128 %0, %1, %2"
                     :: "v"(ldst), "v"(byteoff), "s"(gsrc)
                     : "memory");
    }
}

__device__ __forceinline__ void wait_async_all() {
    asm volatile("s_wait_asynccnt 0x0" ::: "memory");
}

// ---------------------------------------------------------------------------
// Main kernel: one wave32 handles 32 points (two 16-row WMMA A-tiles, two
// independent accumulators so the XDL chain is 2-deep pipelined and every
// B-operand LDS read feeds two V_WMMA_F32_16X16X4_F32 ops).
//   score[n][k] = dot(x_n, c_k) - 0.5*||c_k||^2 ; argmax_k == argmin dist2
// Codebook is double-buffered through LDS via async-to-LDS DMA.
// ---------------------------------------------------------------------------
__global__ __launch_bounds__(256) void vq_wmma_kernel(const float* __restrict__ x,
                                                      const float* __restrict__ cb,
                                                      const float* __restrict__ cbh,
                                                      float* __restrict__ out) {
    extern __shared__ float smem[];                 // 2 x CHUNK_FLOATS (128 KB)

    const int tid  = threadIdx.x;
    const int lane = tid & 31;
    const int wave = tid >> 5;
    const int n0   = (blockIdx.x * WAVES_PER_BLOCK + wave) * ROWS_PER_WAVE;
    const int b    = n0 >> 12;                      // tiles never straddle b (4096%32==0)
    const int l0   = n0 & (SEQ - 1);
    const int half = lane >> 4;                     // 0: K={0,1}; 1: K={2,3}
    const int l16  = lane & 15;

    // ---- A operands, resident for all 2048 WMMAs of this wave ----
    const float* xb = x + (size_t)b * LATENT_DIM * SEQ + (size_t)(l0 + l16);
    v2f a0[16], a1[16];
#pragma unroll
    for (int s = 0; s < 16; ++s) {
        const int c = 4 * s + 2 * half;
        a0[s].x = xb[(size_t)c * SEQ];
        a0[s].y = xb[(size_t)(c + 1) * SEQ];
        a1[s].x = xb[(size_t)c * SEQ + 16];
        a1[s].y = xb[(size_t)(c + 1) * SEQ + 16];
    }

    float best0[8], best1[8];
    int   bidx0[8], bidx1[8];
#pragma unroll
    for (int r = 0; r < 8; ++r) {
        best0[r] = -3.4e38f; bidx0[r] = 0;
        best1[r] = -3.4e38f; bidx1[r] = 0;
    }

    // ---- Stage chunk 0, then chunked double-buffered main loop ----
    issue_chunk_async(cb, smem, tid);
    wait_async_all();
    __syncthreads();

    for (int ch = 0; ch < NCHUNK; ++ch) {
        if (ch + 1 < NCHUNK)                        // prefetch next chunk (overlaps WMMA)
            issue_chunk_async(cb + (size_t)(ch + 1) * CHUNK_FLOATS,
                              smem + (size_t)((ch + 1) & 1) * CHUNK_FLOATS, tid);

        const float* lbuf = smem + (size_t)(ch & 1) * CHUNK_FLOATS;

        for (int kt = 0; kt < CHUNK_K / 16; ++kt) {
            const int kloc  = kt * 16 + l16;        // row within chunk (this lane's column)
            const int kglob = ch * CHUNK_K + kloc;
            const float* brow = lbuf + (size_t)kloc * LATENT_DIM + 2 * half;
            const float negc = -cbh[kglob];         // C-init = -0.5*||c_k||^2 (L2-hot)

            v8f acc0, acc1;
#pragma unroll
            for (int r = 0; r < 8; ++r) { acc0[r] = negc; acc1[r] = negc; }

#pragma unroll
            for (int s = 0; s < 16; ++s) {
                const v2f bv = *(const v2f*)(brow + 4 * s);  // DS_LOAD_B64
                acc0 = __builtin_amdgcn_wmma_f32_16x16x4_f32(
                           false, a0[s], false, bv, (short)0, acc0, false, false);
                acc1 = __builtin_amdgcn_wmma_f32_16x16x4_f32(
                           false, a1[s], false, bv, (short)0, acc1, false, false);
            }

#pragma unroll
            for (int r = 0; r < 8; ++r) {
                if (acc0[r] > best0[r]) { best0[r] = acc0[r]; bidx0[r] = kglob; }
                if (acc1[r] > best1[r]) { best1[r] = acc1[r]; bidx1[r] = kglob; }
            }
        }

        wait_async_all();                           // prefetch landed
        __syncthreads();                            // nobody still reading old buffer
    }

    // ---- Per-row argmax across the 16 lanes of each half ----
#pragma unroll
    for (int r = 0; r < 8; ++r) {
        float v0 = best0[r]; int i0 = bidx0[r];
        float v1 = best1[r]; int i1 = bidx1[r];
#pragma unroll
        for (int off = 8; off >= 1; off >>= 1) {
            const float ov0 = __shfl_xor(v0, off, 32);
            const int   oi0 = __shfl_xor(i0, off, 32);
            const float ov1 = __shfl_xor(v1, off, 32);
            const int   oi1 = __shfl_xor(i1, off, 32);
            if (ov0 > v0 || (ov0 == v0 && oi0 < i0)) { v0 = ov0; i0 = oi0; }
            if (ov1 > v1 || (ov1 == v1 && oi1 < i1)) { v1 = ov1; i1 = oi1; }
        }
        bidx0[r] = i0; bidx1[r] = i1;
    }

    // ---- Broadcast the 32 winning indices to every lane ----
    int rowk[32];
#pragma unroll
    for (int r = 0; r < 8; ++r) {
        rowk[r]      = __shfl(bidx0[r], 0, 32);     // rows 0..7   (tile0, low half)
        rowk[r + 8]  = __shfl(bidx0[r], 16, 32);    // rows 8..15  (tile0, high half)
        rowk[r + 16] = __shfl(bidx1[r], 0, 32);     // rows 16..23 (tile1, low half)
        rowk[r + 24] = __shfl(bidx1[r], 16, 32);    // rows 24..31 (tile1, high half)
    }

    // ---- indices output (section 3), stored as float per harness convention ----
    if (l16 == 0) {
#pragma unroll
        for (int r = 0; r < 8; ++r) {
            out[(size_t)3 * S_OUT + (size_t)n0 + half * 8 + r]      = (float)bidx0[r];
            out[(size_t)3 * S_OUT + (size_t)n0 + 16 + half * 8 + r] = (float)bidx1[r];
        }
    }

    // ---- gather codebook rows; write z_q (section 0) and z (section 1), (b,c,l) ----
    float* zq = out;
    float* zf = out + (size_t)S_OUT;
#pragma unroll
    for (int m = 0; m < 32; ++m) {
        const float* crow = cb + (size_t)rowk[m] * LATENT_DIM;
#pragma unroll
        for (int t = 0; t < 2; ++t) {
            const int c = lane + 32 * t;
            const float v = crow[c];
            const size_t o = (size_t)b * LATENT_DIM * SEQ + (size_t)c * SEQ + (size_t)(l0 + m);
            zq[o] = v;
            zf[o] = v;
        }
    }
}

// ---------------------------------------------------------------------------
extern "C" void kernel_launch(void* const* d_in, const int* in_sizes, int n_in,
                              void* d_out, int out_size, void* d_ws, size_t ws_size,
                              hipStream_t stream) {
    const float* x  = (const float*)d_in[0];   // (32, 64, 4096) fp32
    const float* cb = (const float*)d_in[1];   // (1024, 64) fp32
    float* out = (float*)d_out;
    float* cbh = (float*)d_ws;                 // 1024 floats of scratch

    vq_cbh_kernel<<<1, 1024, 0, stream>>>(cb, cbh);

    const size_t lds_bytes = 2u * CHUNK_FLOATS * sizeof(float);   // 128 KB, 2 blocks/WGP
    vq_wmma_kernel<<<N_TOTAL / ROWS_PER_BLOCK, 256, lds_bytes, stream>>>(x, cb, cbh, out);

    // Section 2 is a verbatim echo of x: graph-capture-safe async D2D copy.
    (void)hipMemcpyAsync(out + (size_t)2 * S_OUT, x, (size_t)S_OUT * sizeof(float),
                         hipMemcpyDeviceToDevice, stream);
}